// SphereSpectraIntegrator_63720134803863
// MI455X (gfx1250) — compile-verified
//
#include <hip/hip_runtime.h>

typedef __attribute__((ext_vector_type(2))) float v2f;
typedef __attribute__((ext_vector_type(8))) float v8f;

#define TWO_OVER_SQRT_PI 1.1283791670955126f
#define CUTOFF2 6.25f   // CUTOFF^2: |arg| <= 2.5  <=>  arg^2 <= 6.25

// Compute masked arg*exp(-arg^2) (A operand) and (2/sqrt(pi))*c^2*w (B operand)
// for one pair of lines held in float2 registers.
__device__ __forceinline__ void epr_pair(float bval, v2f bm, v2f c, v2f w,
                                         v2f& A, v2f& Bw)
{
    const float t0  = (bm.x - bval) * c.x;
    const float t1  = (bm.y - bval) * c.y;
    const float t0s = t0 * t0, t1s = t1 * t1;
    const float e0  = __expf(-t0s);
    const float e1  = __expf(-t1s);
    A.x  = (t0s <= CUTOFF2) ? t0 * e0 : 0.f;
    A.y  = (t1s <= CUTOFF2) ? t1 * e1 : 0.f;
    Bw.x = TWO_OVER_SQRT_PI * c.x * c.x * w.x;
    Bw.y = TWO_OVER_SQRT_PI * c.y * c.y * w.y;
}

// Each wave: 16 spectral fields (matrix M dim) x one chunk of lines (K dim,
// consumed 4 at a time through V_WMMA_F32_16X16X4_F32, 2x unrolled).
__global__ __launch_bounds__(256) void epr_wmma_kernel(
    const float* __restrict__ B_mean,
    const float* __restrict__ c_ext,
    const float* __restrict__ B_val,
    const float* __restrict__ weights,
    float* __restrict__ partial,      // [nChunks][nFields]
    int nLines, int nFields, int nChunks, int linesPerChunk)
{
    // Wave id is uniform across the wave; tell the compiler so loop control,
    // bounds and base addresses live in SGPRs (and EXEC provably stays all-1s
    // around every WMMA, as the ISA requires).
    const int tid  = blockIdx.x * (int)blockDim.x + (int)threadIdx.x;
    const int wave = __builtin_amdgcn_readfirstlane(tid >> 5);
    const int lane = (int)threadIdx.x & 31;
    const int nFieldTiles = nFields >> 4;
    const int ftile = wave % nFieldTiles;
    const int chunk = wave / nFieldTiles;
    if (chunk >= nChunks) return;          // wave-uniform branch

    const int fcol  = lane & 15;           // field within tile (A: M, D: N)
    const int kbase = (lane >> 4) << 1;    // K pair: lanes 0-15 -> K=0,1 ; 16-31 -> K=2,3

    const float bval = B_val[ftile * 16 + fcol];

    const int start = chunk * linesPerChunk;
    int end = start + linesPerChunk;
    if (end > nLines) end = nLines;
    const int span    = end - start;
    const int endFull = start + (span & ~7);   // 8-line (2-WMMA) unguarded body

    v8f acc = {0.f, 0.f, 0.f, 0.f, 0.f, 0.f, 0.f, 0.f};

    int n0 = start;
    for (; n0 < endFull; n0 += 8) {
        const int i0 = n0 + kbase;         // in-bounds by construction (i0+5 < end)
        const v2f bm0 = *(const v2f*)(B_mean  + i0);
        const v2f c0  = *(const v2f*)(c_ext   + i0);
        const v2f w0  = *(const v2f*)(weights + i0);
        const v2f bm1 = *(const v2f*)(B_mean  + i0 + 4);
        const v2f c1  = *(const v2f*)(c_ext   + i0 + 4);
        const v2f w1  = *(const v2f*)(weights + i0 + 4);

        v2f A0, B0, A1, B1;
        epr_pair(bval, bm0, c0, w0, A0, B0);
        epr_pair(bval, bm1, c1, w1, A1, B1);

        acc = __builtin_amdgcn_wmma_f32_16x16x4_f32(false, A0, false, B0,
                                                    (short)0, acc, false, false);
        acc = __builtin_amdgcn_wmma_f32_16x16x4_f32(false, A1, false, B1,
                                                    (short)0, acc, false, false);
    }

    // Guarded 4-line tail (clamped loads keep EXEC untouched; zero weight = no-op line)
    for (; n0 < end; n0 += 4) {
        const int i0 = n0 + kbase;
        const int i1 = i0 + 1;
        const int j0 = (i0 < nLines) ? i0 : (nLines - 1);
        const int j1 = (i1 < nLines) ? i1 : (nLines - 1);
        v2f bm, c, w;
        bm.x = B_mean[j0];  bm.y = B_mean[j1];
        c.x  = c_ext[j0];   c.y  = c_ext[j1];
        w.x  = (i0 < nLines) ? weights[j0] : 0.f;
        w.y  = (i1 < nLines) ? weights[j1] : 0.f;

        v2f A, Bw;
        epr_pair(bval, bm, c, w, A, Bw);
        acc = __builtin_amdgcn_wmma_f32_16x16x4_f32(false, A, false, Bw,
                                                    (short)0, acc, false, false);
    }

    // All 16 columns of D are identical; column 0 lives in lane 0 (M=0..7)
    // and lane 16 (M=8..15), one M per accumulator VGPR.
    if (fcol == 0) {
        const int mbase = (lane >> 4) * 8;
        float* p = partial + (size_t)chunk * nFields + ftile * 16 + mbase;
        #pragma unroll
        for (int i = 0; i < 8; ++i) p[i] = acc[i];
    }
}

__global__ __launch_bounds__(256) void epr_reduce_kernel(
    const float* __restrict__ partial, float* __restrict__ out,
    int nFields, int nChunks)
{
    const int f = blockIdx.x * (int)blockDim.x + (int)threadIdx.x;
    if (f >= nFields) return;
    float s = 0.f;
    for (int c = 0; c < nChunks; ++c)
        s += partial[(size_t)c * nFields + f];
    out[f] = s;
}

extern "C" void kernel_launch(void* const* d_in, const int* in_sizes, int n_in,
                              void* d_out, int out_size, void* d_ws, size_t ws_size,
                              hipStream_t stream) {
    const float* B_mean  = (const float*)d_in[0];
    const float* c_ext   = (const float*)d_in[1];
    const float* B_val   = (const float*)d_in[2];
    const float* weights = (const float*)d_in[3];
    float* out = (float*)d_out;

    const int nLines  = in_sizes[0];
    const int nFields = in_sizes[2];      // 1024, multiple of 16
    const int nFieldTiles = nFields / 16;

    // Split lines into chunks for wave-level parallelism (8192 waves at 128).
    int nChunks = 128;
    while (nChunks > 1 && (size_t)nChunks * nFields * sizeof(float) > ws_size)
        nChunks >>= 1;
    float* partial = (nChunks > 1) ? (float*)d_ws : out;

    int linesPerChunk = ((nLines + nChunks - 1) / nChunks + 3) & ~3;

    const int waves   = nChunks * nFieldTiles;
    const int threads = waves * 32;
    const int block   = 256;
    const int grid    = (threads + block - 1) / block;

    epr_wmma_kernel<<<grid, block, 0, stream>>>(B_mean, c_ext, B_val, weights,
                                                partial, nLines, nFields,
                                                nChunks, linesPerChunk);
    if (nChunks > 1) {
        epr_reduce_kernel<<<(nFields + block - 1) / block, block, 0, stream>>>(
            partial, out, nFields, nChunks);
    }
}